// ParallelConvSSM3D_19628000543416
// MI455X (gfx1250) — compile-verified
//
#include <hip/hip_runtime.h>
#include <hip/hip_bf16.h>

// ---------------------------------------------------------------------------
// ParallelConvSSM3D on MI455X (gfx1250, wave32)
//   x: (B=4, T=32, H=64, W=64, C=64) f32
//   DFT along each axis == small GEMM  F(NxN) x S(Nx64)  (C contiguous)
//   -> executed with V_WMMA_F32_16X16X4_F32 (fp32 precision for FFT),
//      LDS staging via GLOBAL_LOAD_ASYNC_TO_LDS_B128 when available.
// Workspace (floats):
//   [tables 30720][Afre 8M][Afim 8M][Bfre 8M][Bfim 8M][Xre 32M][Xim 32M]
// ---------------------------------------------------------------------------

typedef float v2f __attribute__((ext_vector_type(2)));
typedef float v8f __attribute__((ext_vector_type(8)));
typedef int   v4i __attribute__((ext_vector_type(4)));

#if defined(__has_builtin)
#  if __has_builtin(__builtin_amdgcn_global_load_async_to_lds_b128)
#    define CSSM_ASYNC 1
#  endif
#endif
#ifndef CSSM_ASYNC
#  define CSSM_ASYNC 0
#endif

#define PI_F 3.14159265358979323846f

static constexpr int Bn = 4, Tn = 32, Hn = 64, Wn = 64, Cn = 64;
static constexpr int KT = 5, KS = 7;

// table offsets (floats)
static constexpr long long OF_F64RE = 0,     OF_F64IM = 4096,  OF_NF64IM = 8192;
static constexpr long long OF_IF64RE = 12288, OF_IF64IM = 16384, OF_NIF64IM = 20480;
static constexpr long long OF_FTRE = 24576,  OF_FTIM = 25600,  OF_NFTIM = 26624;
static constexpr long long OF_IFTRE = 27648, OF_IFTIM = 28672, OF_NIFTIM = 29696;
static constexpr long long TABSZ = 30720;

// ---------------------------------------------------------------------------
// K0: DFT matrices. Forward F[m][k] = e^{-2pi i mk/N}; inverse conj/N folded.
// Also store negated-imag copies so the WMMA kernel never needs VALU negation.
// ---------------------------------------------------------------------------
__global__ void cssm_build_tables(float* __restrict__ ws) {
  for (int i = threadIdx.x; i < 4096; i += blockDim.x) {
    int m = i >> 6, k = i & 63;
    float ang = -2.0f * PI_F * (float)((m * k) & 63) * (1.0f / 64.0f);
    float s, c; sincosf(ang, &s, &c);
    ws[OF_F64RE + i] = c;                   ws[OF_F64IM + i] = s;
    ws[OF_NF64IM + i] = -s;
    ws[OF_IF64RE + i] = c * (1.0f/64.0f);   ws[OF_IF64IM + i] = -s * (1.0f/64.0f);
    ws[OF_NIF64IM + i] = s * (1.0f/64.0f);
  }
  for (int i = threadIdx.x; i < 1024; i += blockDim.x) {
    int m = i >> 5, k = i & 31;
    float ang = -2.0f * PI_F * (float)((m * k) & 31) * (1.0f / 32.0f);
    float s, c; sincosf(ang, &s, &c);
    ws[OF_FTRE + i] = c;                    ws[OF_FTIM + i] = s;
    ws[OF_NFTIM + i] = -s;
    ws[OF_IFTRE + i] = c * (1.0f/32.0f);    ws[OF_IFTIM + i] = -s * (1.0f/32.0f);
    ws[OF_NIFTIM + i] = s * (1.0f/32.0f);
  }
}

// ---------------------------------------------------------------------------
// K1: kernels -> frequency domain, evaluated directly from the 245 taps.
// grid: (t,h,wq) = 32*64*16 blocks; block 256 = 64 c  x  4 w
// ---------------------------------------------------------------------------
__global__ __launch_bounds__(256)
void cssm_kernel_freq(const float* __restrict__ Ak, const float* __restrict__ Bk,
                      float* __restrict__ Afre, float* __restrict__ Afim,
                      float* __restrict__ Bfre, float* __restrict__ Bfim) {
  __shared__ float phtr[KT * KS], phti[KT * KS];  // combined (t,h) phases
  __shared__ float pwr[4 * KS],  pwi[4 * KS];     // per-w phases (4 w per block)
  int bi = blockIdx.x;
  int wq = bi & 15;
  int h  = (bi >> 4) & 63;
  int t  = bi >> 10;
  int tid = threadIdx.x;
  if (tid < KT * KS) {
    int kt = tid / KS, kh = tid % KS;
    int pt = (t * (kt - 2)) % 32;
    int ph = (h * (kh - 3)) % 64;
    float ang = -2.0f * PI_F * ((float)pt * (1.0f/32.0f) + (float)ph * (1.0f/64.0f));
    float s, c; sincosf(ang, &s, &c);
    phtr[tid] = c; phti[tid] = s;
  } else if (tid >= 64 && tid < 64 + 4 * KS) {
    int j = tid - 64;
    int wl = j / KS, kw = j % KS;
    int w = wq * 4 + wl;
    int pw = (w * (kw - 3)) % 64;
    float ang = -2.0f * PI_F * (float)pw * (1.0f/64.0f);
    float s, c; sincosf(ang, &s, &c);
    pwr[j] = c; pwi[j] = s;
  }
  __syncthreads();

  int c  = tid & 63;
  int wl = tid >> 6;
  int w  = wq * 4 + wl;
  const float* At = Ak + c * (KT * KS * KS);
  const float* Bt = Bk + c * (KT * KS * KS);
  float aR = 0.f, aI = 0.f, bR = 0.f, bI = 0.f;
  for (int kt = 0; kt < KT; ++kt) {
    for (int kh = 0; kh < KS; ++kh) {
      float pr = phtr[kt * KS + kh], pi = phti[kt * KS + kh];
#pragma unroll
      for (int kw = 0; kw < KS; ++kw) {
        float er = pwr[wl * KS + kw], ei = pwi[wl * KS + kw];
        float rr = pr * er - pi * ei;
        float ri = pr * ei + pi * er;
        float ta = At[(kt * KS + kh) * KS + kw];
        float tb = Bt[(kt * KS + kh) * KS + kw];
        aR += ta * rr; aI += ta * ri;
        bR += tb * rr; bI += tb * ri;
      }
    }
  }
  long long o = (((long long)t * 64 + h) * 64 + w) * 64 + c;
  Afre[o] = aR; Afim[o] = aI; Bfre[o] = bR; Bfim[o] = bI;
}

// ---------------------------------------------------------------------------
// K2: generic in-place complex DFT along one axis via f32 WMMA.
//   D(Nx64) = M(NxN complex) * S(Nx64 complex); one block (8 waves) per slab.
//   Wave -> fixed (component, 16-row band); unconditional WMMA chains
//   (matrix operand chosen by wave-uniform pointer select; negated-imag
//   table precomputed, so no EXEC manipulation anywhere near WMMA).
// ---------------------------------------------------------------------------
template <int N, bool HAS_IM, bool REAL_OUT>
__global__ __launch_bounds__(256)
void cssm_dft_axis(const float* __restrict__ src_re, const float* __restrict__ src_im,
                   float* __restrict__ dst_re, float* __restrict__ dst_im,
                   const float* __restrict__ Mre, const float* __restrict__ Mim,
                   const float* __restrict__ MimNeg,
                   int inner_count, long long outer_stride, long long row_stride) {
  __shared__ float Sre[N][64];
  __shared__ float Sim[N][64];

  int s = blockIdx.x;
  int i = s % inner_count;
  int o = s / inner_count;
  long long base = (long long)o * outer_stride + (long long)i * 64;
  int tid = threadIdx.x;

  // ---- stage slab into LDS (async-to-LDS on CDNA5, float4 fallback) ----
#if CSSM_ASYNC
  for (int j = tid; j < N * 16; j += 256) {
    int r  = j >> 4;
    int c4 = (j & 15) << 2;
    long long g = base + (long long)r * row_stride + c4;
    __builtin_amdgcn_global_load_async_to_lds_b128(
        (v4i*)(src_re + g), (v4i*)&Sre[r][c4], 0, 0);
    if constexpr (HAS_IM) {
      __builtin_amdgcn_global_load_async_to_lds_b128(
          (v4i*)(src_im + g), (v4i*)&Sim[r][c4], 0, 0);
    }
  }
  asm volatile("s_wait_asynccnt 0x0" ::: "memory");
#else
  for (int j = tid; j < N * 16; j += 256) {
    int r  = j >> 4;
    int c4 = (j & 15) << 2;
    long long g = base + (long long)r * row_stride + c4;
    *reinterpret_cast<float4*>(&Sre[r][c4]) =
        *reinterpret_cast<const float4*>(src_re + g);
    if constexpr (HAS_IM) {
      *reinterpret_cast<float4*>(&Sim[r][c4]) =
          *reinterpret_cast<const float4*>(src_im + g);
    }
  }
#endif
  __syncthreads();

  int wv = tid >> 5;
  int lane = tid & 31;
  constexpr int COMPS = REAL_OUT ? 1 : 2;
  constexpr int MBc = N / 16;               // row bands
  constexpr int TILES = COMPS * MBc * 4;    // comps * bands * n-tiles
  constexpr int PW = TILES / 8;             // tiles per wave (>=1)
  int t0 = wv * PW;
  int comp = __builtin_amdgcn_readfirstlane(t0 / (MBc * 4)); // 0=re out, 1=im out
  int mb   = __builtin_amdgcn_readfirstlane((t0 % (MBc * 4)) / 4);
  int nt0  = __builtin_amdgcn_readfirstlane(t0 % 4);

  // A tiles (DFT matrix) for this wave's row band, kept in registers.
  //   complex in : comp0: acc = Mre*Sre + (-Mim)*Sim
  //                comp1: acc = Mre*Sim + ( Mim)*Sre
  //   real in    : comp0: acc = Mre*Sre ; comp1: acc = Mim*Sre
  constexpr int NK = N / 4;
  v2f aA[NK], aB[NK];
  {
    const float* MA = HAS_IM ? Mre : (comp ? Mim : Mre);
    const float* MBm = (comp == 0) ? MimNeg : Mim;
    int row  = mb * 16 + (lane & 15);
    int koff = (lane >= 16) ? 2 : 0;
#pragma unroll
    for (int k = 0; k < NK; ++k) {
      int col = 4 * k + koff;
      v2f a; a.x = MA[row * N + col]; a.y = MA[row * N + col + 1];
      aA[k] = a;
      if constexpr (HAS_IM) {
        v2f b2; b2.x = MBm[row * N + col]; b2.y = MBm[row * N + col + 1];
        aB[k] = b2;
      }
    }
  }

  int koff = (lane >= 16) ? 2 : 0;
  int nl = lane & 15;
  const float(*SaP)[64] = (!HAS_IM || comp == 0) ? Sre : Sim;  // partner of chain A
  const float(*SbP)[64] = (comp == 0) ? Sim : Sre;             // partner of chain B

  for (int q = 0; q < PW; ++q) {
    int nt = nt0 + q;
    int n = nt * 16 + nl;
    v8f acc = {0.f, 0.f, 0.f, 0.f, 0.f, 0.f, 0.f, 0.f};
#pragma unroll
    for (int k = 0; k < NK; ++k) {
      v2f bv; bv.x = SaP[4 * k + koff][n]; bv.y = SaP[4 * k + koff + 1][n];
      acc = __builtin_amdgcn_wmma_f32_16x16x4_f32(
          false, aA[k], false, bv, (short)0, acc, false, false);
      if constexpr (HAS_IM) {
        v2f bw; bw.x = SbP[4 * k + koff][n]; bw.y = SbP[4 * k + koff + 1][n];
        acc = __builtin_amdgcn_wmma_f32_16x16x4_f32(
            false, aB[k], false, bw, (short)0, acc, false, false);
      }
    }
    float* dst = comp ? dst_im : dst_re;
    int mhi = (lane >= 16) ? 8 : 0;
#pragma unroll
    for (int r = 0; r < 8; ++r) {
      int m = mb * 16 + r + mhi;
      dst[base + (long long)m * row_stride + n] = acc[r];
    }
  }
}

// ---------------------------------------------------------------------------
// K3: first-order recurrence along (frequency) T axis, in place over x_f.
//   h_t = a_t * h_{t-1} + (B_f * x_f)_t   (complex), h_{-1}=0
// ---------------------------------------------------------------------------
__global__ __launch_bounds__(256)
void cssm_scan_t(float* __restrict__ Xre, float* __restrict__ Xim,
                 const float* __restrict__ Afre, const float* __restrict__ Afim,
                 const float* __restrict__ Bfre, const float* __restrict__ Bfim) {
  int gid = blockIdx.x * 256 + threadIdx.x;
  const int HWC = Hn * Wn * Cn;  // 262144
  int hwc = gid % HWC;
  int b = gid / HWC;
  long long xbase = (long long)b * (Tn * HWC) + hwc;
  float hre = 0.f, him = 0.f;
  for (int t = 0; t < Tn; ++t) {
    long long xi = xbase + (long long)t * HWC;
    long long ki = (long long)hwc + (long long)t * HWC;
    float xr = Xre[xi], xm = Xim[xi];
    float br = Bfre[ki], bimv = Bfim[ki];
    float ar = Afre[ki], ai = Afim[ki];
    float vr = br * xr - bimv * xm;
    float vi = br * xm + bimv * xr;
    float nr = ar * hre - ai * him + vr;
    float nm = ar * him + ai * hre + vi;
    hre = nr; him = nm;
    Xre[xi] = hre; Xim[xi] = him;
  }
}

// ---------------------------------------------------------------------------
extern "C" void kernel_launch(void* const* d_in, const int* in_sizes, int n_in,
                              void* d_out, int out_size, void* d_ws, size_t ws_size,
                              hipStream_t stream) {
  const float* x  = (const float*)d_in[0];
  const float* Ak = (const float*)d_in[1];
  const float* Bk = (const float*)d_in[2];
  float* out = (float*)d_out;
  float* ws  = (float*)d_ws;

  // workspace carve-up (floats)
  const long long KPL = 8388608LL;    // 32*64*64*64
  const long long XPL = 33554432LL;   // 4*32*64*64*64
  float* Afre = ws + TABSZ;
  float* Afim = Afre + KPL;
  float* Bfre = Afim + KPL;
  float* Bfim = Bfre + KPL;
  float* Xre  = Bfim + KPL;
  float* Xim  = Xre + XPL;
  const float* F64re = ws + OF_F64RE,  *F64im = ws + OF_F64IM,  *nF64im = ws + OF_NF64IM;
  const float* iF64re = ws + OF_IF64RE, *iF64im = ws + OF_IF64IM, *niF64im = ws + OF_NIF64IM;
  const float* Ftre = ws + OF_FTRE,    *Ftim = ws + OF_FTIM,    *nFtim = ws + OF_NFTIM;
  const float* iFtre = ws + OF_IFTRE,  *iFtim = ws + OF_IFTIM,  *niFtim = ws + OF_NIFTIM;

  cssm_build_tables<<<1, 256, 0, stream>>>(ws);
  cssm_kernel_freq<<<32 * 64 * 16, 256, 0, stream>>>(Ak, Bk, Afre, Afim, Bfre, Bfim);

  // forward FFT over W (real input), H, T   -- in place on (Xre,Xim)
  cssm_dft_axis<64, false, false><<<Bn * Tn * Hn, 256, 0, stream>>>(
      x, nullptr, Xre, Xim, F64re, F64im, nF64im, 1, 4096LL, 64LL);
  cssm_dft_axis<64, true, false><<<Bn * Tn * Wn, 256, 0, stream>>>(
      Xre, Xim, Xre, Xim, F64re, F64im, nF64im, 64, 262144LL, 4096LL);
  cssm_dft_axis<32, true, false><<<Bn * Hn * Wn, 256, 0, stream>>>(
      Xre, Xim, Xre, Xim, Ftre, Ftim, nFtim, 4096, 8388608LL, 262144LL);

  // SSM recurrence along frequency-T
  cssm_scan_t<<<4096, 256, 0, stream>>>(Xre, Xim, Afre, Afim, Bfre, Bfim);

  // inverse FFT over T, H, W (real output)
  cssm_dft_axis<32, true, false><<<Bn * Hn * Wn, 256, 0, stream>>>(
      Xre, Xim, Xre, Xim, iFtre, iFtim, niFtim, 4096, 8388608LL, 262144LL);
  cssm_dft_axis<64, true, false><<<Bn * Tn * Wn, 256, 0, stream>>>(
      Xre, Xim, Xre, Xim, iF64re, iF64im, niF64im, 64, 262144LL, 4096LL);
  cssm_dft_axis<64, true, true><<<Bn * Tn * Hn, 256, 0, stream>>>(
      Xre, Xim, out, nullptr, iF64re, iF64im, niF64im, 1, 4096LL, 64LL);
}